// VectorQuantizer_50869592653967
// MI455X (gfx1250) — compile-verified
//
#include <hip/hip_runtime.h>

typedef __attribute__((ext_vector_type(2))) float v2f;
typedef __attribute__((ext_vector_type(4))) float v4f;
typedef __attribute__((ext_vector_type(8))) float v8f;
typedef __attribute__((ext_vector_type(4))) unsigned int u32x4;
typedef __attribute__((ext_vector_type(4))) int i32x4;
typedef __attribute__((ext_vector_type(8))) int i32x8;

#define Bb    32
#define Dd    256
#define HWs   1024             // H*W
#define Kk    1024
#define Nn    (Bb * HWs)       // 32768 latent rows
#define STX   260              // padded LDS stride (floats): 260 % 64 == 4 -> conflict-free b64 reads
#define STE   260
#define ROWS  32               // latent rows per block (2 row tiles)
#define KTILE 64               // codebook rows staged per iteration (4 col tiles)

#if defined(__has_builtin)
#if __has_builtin(__builtin_amdgcn_tensor_load_to_lds) && __has_builtin(__builtin_amdgcn_s_wait_tensorcnt)
#define HAVE_TDM 1
#endif
#endif

// ---------------------------------------------------------------------------
// Kernel 1: enorm[k] = ||e_k||^2 ; zero the loss accumulator ws[Kk].
// ---------------------------------------------------------------------------
__global__ __launch_bounds__(1024) void vq_prep(const float* __restrict__ emb,
                                                float* __restrict__ ws) {
  const int k = threadIdx.x;
  const v4f* row = (const v4f*)(emb + (size_t)k * Dd);
  float s = 0.f;
#pragma unroll 8
  for (int i = 0; i < Dd / 4; ++i) {
    v4f v = row[i];
    s += v.x * v.x + v.y * v.y + v.z * v.z + v.w * v.w;
  }
  ws[k] = s;
  if (k == 0) ws[Kk] = 0.f;
}

// ---------------------------------------------------------------------------
// Kernel 2: fused GEMM (v_wmma_f32_16x16x4_f32) + argmin + gather + loss.
// 256 threads = 8 waves; wave w: rowTile rt = w>>2, colTile ct = w&3.
// Block covers ROWS=32 latent rows, loops K in KTILE=64 column tiles.
// E tiles arrive via the Tensor Data Mover (async DMA into padded LDS).
// ---------------------------------------------------------------------------
__global__ __launch_bounds__(256) void vq_main(const float* __restrict__ lat,
                                               const float* __restrict__ emb,
                                               const float* __restrict__ enorm,
                                               float* __restrict__ wacc,
                                               float* __restrict__ out) {
  __shared__ float lds_x[ROWS * STX];        // 32 x 256 latent tile (padded)
  __shared__ float lds_e[KTILE * STE];       // 64 x 256 codebook tile (padded)
  __shared__ float lds_en[Kk];               // all 1024 code norms
  __shared__ float red_val[8][16];
  __shared__ int   red_idx[8][16];
  __shared__ int   fin_idx[ROWS];

  const int tid  = threadIdx.x;
  const int w    = tid >> 5;                 // wave 0..7
  const int lane = tid & 31;
  const int rt   = w >> 2;                   // row tile 0..1
  const int ct   = w & 3;                    // col tile 0..3
  const int n0   = blockIdx.x * ROWS;
  const int b    = n0 >> 10;
  const int hw0  = n0 & (HWs - 1);
  const float* xbase = lat + (size_t)b * Dd * HWs + hw0;   // + d*HWs + r

  // ---- stage X tile (transpose [D][HW] -> [row][d]) and code norms ----
  {
    const int r2 = tid & 31;
    const int d0 = tid >> 5;                 // 0..7
#pragma unroll
    for (int j = 0; j < 32; ++j) {
      const int d = d0 + j * 8;
      lds_x[r2 * STX + d] = xbase[(size_t)d * HWs + r2];  // consecutive tid -> coalesced
    }
#pragma unroll
    for (int i = 0; i < 4; ++i) lds_en[tid + i * 256] = enorm[tid + i * 256];
  }

  const int m    = lane & 15;                // A row / B column within tile
  const int koff = (lane >> 4) * 2;          // f32 WMMA K sub-offset

  float bestVal[8];
  int   bestIdx[8];
#pragma unroll
  for (int i = 0; i < 8; ++i) { bestVal[i] = -3.402823466e38f; bestIdx[i] = 0; }

  for (int kb = 0; kb < Kk / KTILE; ++kb) {
    __syncthreads();                         // lds_e free to overwrite; lds_x visible

#if defined(HAVE_TDM)
    if (w == 0) {
      // Tensor DMA: 2-D tile 256 x 64 of 4-byte elements, row stride 256,
      // LDS padding 4 DWORDs every 256 DWORDs -> 260-float padded rows.
      const unsigned long long ga =
          (unsigned long long)(uintptr_t)(emb + (size_t)kb * KTILE * Dd);
      const unsigned lds_off = (unsigned)(uintptr_t)(&lds_e[0]);
      u32x4 g0 = { 1u,                                   // count=1, user descriptor
                   lds_off,                              // lds_addr
                   (unsigned)ga,                         // global_addr[31:0]
                   ((unsigned)(ga >> 32) & 0x01FFFFFFu) | 0x80000000u }; // type=2
      i32x8 g1 = { (int)((2u << 16)     // data_size = 4B
                       | (1u << 20)     // pad_enable
                       | (7u << 22)     // pad_interval = 256 DWORDs
                       | (3u << 25)),   // pad_amount   = 4 DWORDs
                   (int)(256u << 16),   // tensor_dim0[15:0] = 256 (bits 79:48)
                   (int)(64u << 16),    // tensor_dim1[15:0] = 64  (bits 111:80)
                   (int)(256u << 16),   // tile_dim0 = 256         (bits 127:112)
                   (int)64,             // tile_dim1 = 64          (bits 143:128)
                   (int)256,            // tensor_dim0_stride = 256(bits 207:160)
                   0, 0 };
      i32x4 gz = { 0, 0, 0, 0 };
#if __clang_major__ >= 23
      i32x8 gz8 = { 0, 0, 0, 0, 0, 0, 0, 0 };
      __builtin_amdgcn_tensor_load_to_lds(g0, g1, gz, gz, gz8, 0);
#else
      __builtin_amdgcn_tensor_load_to_lds(g0, g1, gz, gz, 0);
#endif
      __builtin_amdgcn_s_wait_tensorcnt(0);
      asm volatile("" ::: "memory");
    }
#else
    {
      const v4f* src = (const v4f*)(emb + (size_t)kb * KTILE * Dd);
#pragma unroll
      for (int i = 0; i < 16; ++i) {
        const int f4  = tid + i * 256;       // 0..4095
        const int row = f4 >> 6;
        const int c4  = f4 & 63;
        *(v4f*)&lds_e[row * STE + c4 * 4] = src[f4];
      }
    }
#endif
    __syncthreads();

    // ---- 16x16 score tile; two independent accumulator chains for ILP ----
    v8f c0 = {0.f, 0.f, 0.f, 0.f, 0.f, 0.f, 0.f, 0.f};
    v8f c1 = {0.f, 0.f, 0.f, 0.f, 0.f, 0.f, 0.f, 0.f};
    const float* xa = &lds_x[(rt * 16 + m) * STX + koff];
    const float* eb = &lds_e[(ct * 16 + m) * STE + koff];
#pragma unroll 8
    for (int d = 0; d < Dd; d += 8) {
      v2f a0 = *(const v2f*)(xa + d);
      v2f b0 = *(const v2f*)(eb + d);
      c0 = __builtin_amdgcn_wmma_f32_16x16x4_f32(false, a0, false, b0,
                                                 (short)0, c0, false, false);
      v2f a1 = *(const v2f*)(xa + d + 4);
      v2f b1 = *(const v2f*)(eb + d + 4);
      c1 = __builtin_amdgcn_wmma_f32_16x16x4_f32(false, a1, false, b1,
                                                 (short)0, c1, false, false);
    }

    // ---- bias with -||e||^2 ; update running argmax of (2*s - enorm) ----
    const int   k  = kb * KTILE + ct * 16 + m;
    const float en = lds_en[k];
#pragma unroll
    for (int i = 0; i < 8; ++i) {
      const float val = 2.f * (c0[i] + c1[i]) - en;
      if (val > bestVal[i]) { bestVal[i] = val; bestIdx[i] = k; }
    }
  }

  // ---- reduce across 16-lane column groups (rows 0-7 / 8-15 of the tile) ----
#pragma unroll
  for (int off = 8; off >= 1; off >>= 1) {
#pragma unroll
    for (int i = 0; i < 8; ++i) {
      const float ov = __shfl_xor(bestVal[i], off, 32);
      const int   oi = __shfl_xor(bestIdx[i], off, 32);
      if (ov > bestVal[i] || (ov == bestVal[i] && oi < bestIdx[i])) {
        bestVal[i] = ov; bestIdx[i] = oi;
      }
    }
  }
  if ((lane & 15) == 0) {
    const int rbase = (lane >> 4) * 8;
#pragma unroll
    for (int i = 0; i < 8; ++i) {
      red_val[w][rbase + i] = bestVal[i];
      red_idx[w][rbase + i] = bestIdx[i];
    }
  }
  __syncthreads();
  if (tid < ROWS) {                          // combine 4 column-tile waves per row
    const int rrt = tid >> 4;
    const int r   = tid & 15;
    float bv = red_val[rrt * 4][r];
    int   bi = red_idx[rrt * 4][r];
#pragma unroll
    for (int cc = 1; cc < 4; ++cc) {
      const float v  = red_val[rrt * 4 + cc][r];
      const int   ii = red_idx[rrt * 4 + cc][r];
      if (v > bv || (v == bv && ii < bi)) { bv = v; bi = ii; }
    }
    fin_idx[tid] = bi;
  }
  __syncthreads();

  // ---- gather winning code rows to output + accumulate (q - x)^2 ----
  {
    const int r2 = tid & 31;
    const int d0 = tid >> 5;                 // 0..7
    const int qi = fin_idx[r2];
    const float* erow  = emb + (size_t)qi * Dd;
    float* obase = out + (size_t)b * Dd * HWs + hw0 + r2;
    float lsum = 0.f;
#pragma unroll
    for (int j = 0; j < 32; ++j) {
      const int d = d0 + j * 8;
      const float e = erow[d];
      const float x = lds_x[r2 * STX + d];
      obase[(size_t)d * HWs] = e;            // consecutive r2 -> coalesced 128B
      const float df = e - x;
      lsum += df * df;
    }
    atomicAdd(wacc, lsum);
  }
}

// ---------------------------------------------------------------------------
// Kernel 3: finalize scalar loss = 1.25 * sum / (N*D)
// ---------------------------------------------------------------------------
__global__ void vq_fin(const float* __restrict__ wacc, float* __restrict__ loss) {
  *loss = (*wacc) * (1.25f / (float)((size_t)Nn * Dd));
}

extern "C" void kernel_launch(void* const* d_in, const int* in_sizes, int n_in,
                              void* d_out, int out_size, void* d_ws, size_t ws_size,
                              hipStream_t stream) {
  const float* lat = (const float*)d_in[0];  // [32, 256, 32, 32]
  const float* emb = (const float*)d_in[1];  // [1024, 256]
  float* out = (float*)d_out;                // [32,256,32,32] then scalar loss
  float* ws  = (float*)d_ws;                 // [0..1023]=enorm, [1024]=loss acc

  vq_prep<<<1, 1024, 0, stream>>>(emb, ws);
  vq_main<<<Nn / ROWS, 256, 0, stream>>>(lat, emb, ws, ws + Kk, out);
  vq_fin<<<1, 1, 0, stream>>>(ws + Kk, out + (size_t)Nn * Dd);
}